// LocalCorrClosedForm_85023172592493
// MI455X (gfx1250) — compile-verified
//
#include <hip/hip_runtime.h>

typedef __attribute__((ext_vector_type(2))) float v2f;
typedef __attribute__((ext_vector_type(8))) float v8f;

#define S2    81
#define CCH   256
#define BATCH 128
#define NBINS 10

// ---------------------------------------------------------------------------
// Kernel 1: per-batch Gram matrix  G = X^T X / C,  X = feat[b] as [256 x 81].
// 288 threads = 9 wave32s, 4 WMMA tiles each (6x6 grid of 16x16 tiles).
// K=256 in 8 chunks of 32; each chunk staged into LDS (zero-padded to 96
// columns) with global_load_async_to_lds_b32, then consumed as aligned
// ds_load_b64 pairs feeding V_WMMA_F32_16X16X4_F32.
// ---------------------------------------------------------------------------
__global__ __launch_bounds__(288) void gram_kernel(const float* __restrict__ feat,
                                                   float* __restrict__ G) {
  const int b    = blockIdx.x;
  const int tid  = threadIdx.x;
  const int wave = tid >> 5;
  const int lane = tid & 31;
  const int hi   = lane >> 4;   // lane half: K pair select / M+8 on output
  const int mn   = lane & 15;   // M for A, N for B
  const float* X  = feat + (size_t)b * CCH * S2;
  float*       Gb = G    + (size_t)b * S2 * S2;

  // Tt[col 0..95][k-in-chunk 0..31], rows padded to 36 floats (144B, 8B-aligned)
  __shared__ float Tt[96][36];

  v8f acc[4] = {{}, {}, {}, {}};

  for (int kc = 0; kc < CCH; kc += 32) {
    // ---- stage chunk: X[kc..kc+31][0..80] -> Tt (cols 81..95 zero) --------
    for (int idx = tid; idx < 96 * 32; idx += 288) {
      const int col = idx >> 5;   // 0..95 (position)
      const int row = idx & 31;   // 0..31 (k within chunk)
      if (col < S2) {
        const float* src = X + (size_t)(kc + row) * S2 + col;
        const unsigned lds_off = (unsigned)(size_t)&Tt[col][row];
        asm volatile("global_load_async_to_lds_b32 %0, %1, off"
                     :: "v"(lds_off),
                        "v"((unsigned long long)(size_t)src)
                     : "memory");
      } else {
        Tt[col][row] = 0.0f;
      }
    }
    asm volatile("s_wait_asynccnt 0x0" ::: "memory");
    __syncthreads();

    // ---- consume: 4 tiles per wave, 8 WMMA per tile per chunk -------------
#pragma unroll
    for (int tt = 0; tt < 4; ++tt) {
      const int t = wave * 4 + tt;        // 0..35
      const int i = (t / 6) * 16 + mn;    // A column of X == G row
      const int j = (t % 6) * 16 + mn;    // B column of X == G col
      v8f c = acc[tt];
#pragma unroll
      for (int kk = 0; kk < 32; kk += 4) {
        const int k0 = kk + 2 * hi;                       // even -> 8B aligned
        const v2f a  = *(const v2f*)&Tt[i][k0];
        const v2f bb = *(const v2f*)&Tt[j][k0];
        c = __builtin_amdgcn_wmma_f32_16x16x4_f32(false, a, false, bb,
                                                  (short)0, c, false, false);
      }
      acc[tt] = c;
    }
    __syncthreads();
  }

  const float scale = 1.0f / (float)CCH;
#pragma unroll
  for (int tt = 0; tt < 4; ++tt) {
    const int t  = wave * 4 + tt;
    const int i0 = (t / 6) * 16;
    const int col = (t % 6) * 16 + mn;
#pragma unroll
    for (int r = 0; r < 8; ++r) {
      const int row = i0 + r + 8 * hi;
      if (row < S2 && col < S2) Gb[row * S2 + col] = acc[tt][r] * scale;
    }
  }
}

// ---------------------------------------------------------------------------
// Kernel 2: one (batch, position) 81x81 SPD system per block.
// Lower triangle gathered from G, +reg*I, in-LDS Cholesky, two tri-solves
// against the distance-map label, factors written to F[b][p][0..80].
// ---------------------------------------------------------------------------
__global__ __launch_bounds__(128) void chol_solve_kernel(const float* __restrict__ G,
                                                         const float* __restrict__ filter_reg,
                                                         const float* __restrict__ gauss,
                                                         float* __restrict__ F) {
  const int p   = blockIdx.x;   // spatial position 0..80
  const int b   = blockIdx.y;   // batch
  const int tid = threadIdx.x;
  __shared__ float A[S2][84];
  __shared__ float yv[S2];
  const float* Gb = G + (size_t)b * S2 * S2;

  const int ph = p / 9, pw = p % 9;
  const float fr  = filter_reg[0];
  const float reg = fmaxf(fr * fr, 1e-10f);

  // Assemble lower triangle: A[d][e] = G[p+off(e)+off(d), p+off(e)] (masked)
  for (int idx = tid; idx < S2 * S2; idx += 128) {
    const int d = idx / S2, e = idx % S2;
    if (d < e) continue;  // Cholesky reads lower triangle only
    const int ey = e / 9 - 4, ex = e % 9 - 4;
    const int qh = ph + ey, qw = pw + ex;
    float val = 0.0f;
    if (qh >= 0 && qh < 9 && qw >= 0 && qw < 9) {
      const int dy = d / 9 - 4, dx = d % 9 - 4;
      const int rh = qh + dy, rw = qw + dx;
      if (rh >= 0 && rh < 9 && rw >= 0 && rw < 9)
        val = Gb[(rh * 9 + rw) * S2 + (qh * 9 + qw)];
    }
    if (d == e) val += reg;
    A[d][e] = val;
  }
  // Label map (DistanceMap -> 1x1 conv with gauss_weight)
  if (tid < S2) {
    const float ky = (float)(tid / 9) - 4.0f;
    const float kx = (float)(tid % 9) - 4.0f;
    const float dist = sqrtf(ky * ky + kx * kx);
    float lv = 0.0f;
    for (int k = 0; k < NBINS; ++k) {
      const float bd = dist * 2.0f - (float)k;   // dist/0.5 - k
      float bv;
      if (k < NBINS - 1) bv = fmaxf(1.0f - fabsf(bd), 0.0f);
      else               bv = fminf(fmaxf(1.0f + bd, 0.0f), 1.0f);
      lv += bv * gauss[k];
    }
    yv[tid] = lv;
  }
  __syncthreads();

  // Right-looking Cholesky, trailing update parallel over columns
  for (int k = 0; k < S2; ++k) {
    if (tid == 0) A[k][k] = sqrtf(A[k][k]);
    __syncthreads();
    if (tid > k && tid < S2) A[tid][k] /= A[k][k];
    __syncthreads();
    if (tid > k && tid < S2) {
      const float ljk = A[tid][k];
      for (int i = tid; i < S2; ++i) A[i][tid] -= A[i][k] * ljk;
    }
    __syncthreads();
  }

  // Forward solve  L y = l   (column-oriented)
  for (int jj = 0; jj < S2; ++jj) {
    if (tid == 0) yv[jj] /= A[jj][jj];
    __syncthreads();
    if (tid > jj && tid < S2) yv[tid] -= A[tid][jj] * yv[jj];
    __syncthreads();
  }
  // Backward solve L^T f = y (reads lower triangle transposed)
  for (int jj = S2 - 1; jj >= 0; --jj) {
    if (tid == 0) yv[jj] /= A[jj][jj];
    __syncthreads();
    if (tid < jj) yv[tid] -= A[jj][tid] * yv[jj];
    __syncthreads();
  }
  if (tid < S2) F[((size_t)b * S2 + p) * S2 + tid] = yv[tid];
}

// ---------------------------------------------------------------------------
// Kernel 3: adjoint correlation as per-batch GEMM  out[b] = X * M^T with
//   M[p,q] = F[b][p][(qh-ph+4)*9 + (qw-pw+4)]  when |dq|<=4 else 0.
// M^T is materialized once per block in zero-padded LDS (96 x 88) so WMMA
// B operands are unconditional aligned ds_load_b64s. 8 waves x 12 tiles.
// ---------------------------------------------------------------------------
__global__ __launch_bounds__(256) void apply_kernel(const float* __restrict__ feat,
                                                    const float* __restrict__ F,
                                                    float* __restrict__ out) {
  const int b    = blockIdx.x;
  const int tid  = threadIdx.x;
  const int wave = tid >> 5;
  const int lane = tid & 31;
  const int hi   = lane >> 4;
  const int mn   = lane & 15;
  const float* X  = feat + (size_t)b * CCH * S2;
  const float* Fb = F    + (size_t)b * S2 * S2;
  float*       Ob = out  + (size_t)b * CCH * S2;

  // Mt[p 0..95][k 0..87]: M[p][k], zero padded; row stride 88*4=352B (8B ok)
  __shared__ float Mt[96][88];
  for (int idx = tid; idx < 96 * 88; idx += 256) {
    const int p = idx / 88;
    const int k = idx - p * 88;
    float val = 0.0f;
    if (p < S2 && k < S2) {
      const int dy = k / 9 - p / 9;
      const int dx = k % 9 - p % 9;
      if (dy >= -4 && dy <= 4 && dx >= -4 && dx <= 4)
        val = Fb[p * S2 + (dy + 4) * 9 + (dx + 4)];
    }
    Mt[p][k] = val;
  }
  __syncthreads();

  for (int t = wave; t < 96; t += 8) {
    const int c = (t / 6) * 16 + mn;   // channel (always < 256)
    const int p = (t % 6) * 16 + mn;   // output position (Mt padded to 96)
    const float* Xc = X + (size_t)c * S2;
    v8f acc = {};
    // main K loop: k0 <= 79 -> feat reads always in-bounds, branchless
    for (int kc = 0; kc < 80; kc += 4) {
      const int k0 = kc + 2 * hi;
      v2f a;
      a.x = Xc[k0];
      a.y = Xc[k0 + 1];
      const v2f bb = *(const v2f*)&Mt[p][k0];
      acc = __builtin_amdgcn_wmma_f32_16x16x4_f32(false, a, false, bb,
                                                  (short)0, acc, false, false);
    }
    {   // tail chunk kc=80: clamp address, mask value (branchless cndmask)
      const int k0 = 80 + 2 * hi;                 // 80 or 82
      const float t0 = Xc[k0 < S2 ? k0 : S2 - 1];
      v2f a;
      a.x = (k0 < S2) ? t0 : 0.0f;
      a.y = 0.0f;                                  // k0+1 in {81,83}: OOB
      const v2f bb = *(const v2f*)&Mt[p][k0];
      acc = __builtin_amdgcn_wmma_f32_16x16x4_f32(false, a, false, bb,
                                                  (short)0, acc, false, false);
    }
#pragma unroll
    for (int r = 0; r < 8; ++r) {
      const int row = (t / 6) * 16 + r + 8 * hi;   // channel
      const int col = (t % 6) * 16 + mn;           // position
      if (col < S2) Ob[(size_t)row * S2 + col] = acc[r];
    }
  }
}

// ---------------------------------------------------------------------------
extern "C" void kernel_launch(void* const* d_in, const int* in_sizes, int n_in,
                              void* d_out, int out_size, void* d_ws, size_t ws_size,
                              hipStream_t stream) {
  const float* feat       = (const float*)d_in[0];
  const float* filter_reg = (const float*)d_in[1];
  const float* gauss      = (const float*)d_in[2];
  float* out = (float*)d_out;

  float* G = (float*)d_ws;                     // 128 * 81 * 81 floats
  float* F = G + (size_t)BATCH * S2 * S2;      // 128 * 81 * 81 floats

  gram_kernel<<<BATCH, 288, 0, stream>>>(feat, G);
  chol_solve_kernel<<<dim3(S2, BATCH), 128, 0, stream>>>(G, filter_reg, gauss, F);
  apply_kernel<<<BATCH, 256, 0, stream>>>(feat, F, out);
}